// GCNNet_55731495633048
// MI455X (gfx1250) — compile-verified
//
#include <hip/hip_runtime.h>
#include <hip/hip_bf16.h>

// ---------------------------------------------------------------------------
// GCN (2 layers, HDIM=128) for gfx1250 using V_WMMA_F32_16X16X4_F32.
// fp32 throughout (workload is L2/scatter-bound, not compute-bound).
// ---------------------------------------------------------------------------

typedef float v2f __attribute__((ext_vector_type(2)));
typedef float v8f __attribute__((ext_vector_type(8)));

#define HDIM 128
#define LDSW 130                 // padded LDS row stride (floats) -> bank-conflict free
#define ROWS_PER_WAVE 16
#define WAVES_PER_BLOCK 4
#define ROWS_PER_BLOCK (ROWS_PER_WAVE * WAVES_PER_BLOCK)
#define LN_EPS 1e-6f

// ---- 16x128 = (16x128) x (128x128) GEMM for one wave -----------------------
// ast: 16 x 128 A block in LDS (row-major, stride LDSW)
// wl : 128 x 128 weight W in LDS (row-major [out_col][k], stride LDSW)
// Computes acc[t][r] = sum_k A[m][k] * W[n][k]  (i.e. A @ W^T)
//   with m = r + 8*(lane>=16), n = 16*t + (lane&15)  (WMMA C/D layout)
__device__ __forceinline__ void gemm16x128(const float* __restrict__ ast,
                                           const float* __restrict__ wl,
                                           int lane, v8f acc[8]) {
  const int hi = lane >> 4;
  const int lo = lane & 15;
  const float* arow = ast + lo * LDSW;     // A: lane = row M
#pragma unroll
  for (int t = 0; t < 8; ++t) {
    const float* bcol = wl + (lo + 16 * t) * LDSW;   // B: lane = col N, B[k][n]=W[n][k]
    v8f c = {0.f, 0.f, 0.f, 0.f, 0.f, 0.f, 0.f, 0.f};
#pragma unroll
    for (int ks = 0; ks < 32; ++ks) {
      const int kb = 4 * ks + 2 * hi;      // ISA A/B 16x4 layout: hi lanes K+2
      v2f a, b;
      a.x = arow[kb]; a.y = arow[kb + 1];
      b.x = bcol[kb]; b.y = bcol[kb + 1];
      // (neg_a, A, neg_b, B, c_mod, C, reuse_a, reuse_b)
      c = __builtin_amdgcn_wmma_f32_16x16x4_f32(false, a, false, b, (short)0, c,
                                                false, false);
    }
    acc[t] = c;
  }
}

// cooperative load of a 128x128 weight matrix into padded LDS
__device__ __forceinline__ void load_weight_lds(float* wl,
                                                const float* __restrict__ W,
                                                int tid, int nthreads) {
  for (int idx = tid; idx < HDIM * (HDIM / 4); idx += nthreads) {
    const int r = idx >> 5;        // row 0..127
    const int c4 = idx & 31;       // float4 index 0..31
    const float4 v = reinterpret_cast<const float4*>(W + r * HDIM)[c4];
    float* p = wl + r * LDSW + c4 * 4;
    p[0] = v.x; p[1] = v.y; p[2] = v.z; p[3] = v.w;
  }
}

// ---- FF layer: ff = LN( relu(X@W1^T+b1)@W2^T + b2 + X ) --------------------
__global__ __launch_bounds__(WAVES_PER_BLOCK * 32)
void ff_layer_kernel(const float* __restrict__ X,
                     const float* __restrict__ W1, const float* __restrict__ b1,
                     const float* __restrict__ W2, const float* __restrict__ b2,
                     const float* __restrict__ lns, const float* __restrict__ lnb,
                     float* __restrict__ ffout, int nNodes) {
  __shared__ float wl[HDIM * LDSW];
  __shared__ float stage[WAVES_PER_BLOCK * ROWS_PER_WAVE * LDSW];

  const int tid = threadIdx.x;
  const int lane = tid & 31;
  const int wv = tid >> 5;
  const int lo = lane & 15;
  const int hi = lane >> 4;
  float* ast = stage + wv * ROWS_PER_WAVE * LDSW;
  const int base = blockIdx.x * ROWS_PER_BLOCK + wv * ROWS_PER_WAVE;

  // stage this wave's 16 X rows (zero-padded past nNodes)
  for (int idx = lane; idx < ROWS_PER_WAVE * (HDIM / 4); idx += 32) {
    const int r = idx >> 5;
    const int c4 = idx & 31;
    const int node = base + r;
    float4 v = make_float4(0.f, 0.f, 0.f, 0.f);
    if (node < nNodes)
      v = reinterpret_cast<const float4*>(X + (size_t)node * HDIM)[c4];
    float* p = ast + r * LDSW + c4 * 4;
    p[0] = v.x; p[1] = v.y; p[2] = v.z; p[3] = v.w;
  }
  load_weight_lds(wl, W1, tid, blockDim.x);
  __syncthreads();

  v8f acc[8];
  gemm16x128(ast, wl, lane, acc);          // X @ W1^T

  __syncthreads();                         // everyone done reading W1
  load_weight_lds(wl, W2, tid, blockDim.x);
  // h = relu(acc + b1) -> back into this wave's stage buffer (A for GEMM2)
#pragma unroll
  for (int t = 0; t < 8; ++t) {
    const int n = 16 * t + lo;
    const float bb = b1[n];
#pragma unroll
    for (int r = 0; r < 8; ++r) {
      const int m = r + 8 * hi;
      ast[m * LDSW + n] = fmaxf(acc[t][r] + bb, 0.f);
    }
  }
  __syncthreads();

  gemm16x128(ast, wl, lane, acc);          // h @ W2^T

  // residual + LayerNorm (row-wise over 128, 16-lane shfl_xor reduction)
  float val[8][8];
#pragma unroll
  for (int t = 0; t < 8; ++t) {
    const int n = 16 * t + lo;
    const float bb = b2[n];
#pragma unroll
    for (int r = 0; r < 8; ++r) {
      const int m = r + 8 * hi;
      const int node = base + m;
      const float xv = (node < nNodes) ? X[(size_t)node * HDIM + n] : 0.f;
      val[t][r] = acc[t][r] + bb + xv;
    }
  }
  float s1[8], s2[8];
#pragma unroll
  for (int r = 0; r < 8; ++r) { s1[r] = 0.f; s2[r] = 0.f; }
#pragma unroll
  for (int t = 0; t < 8; ++t)
#pragma unroll
    for (int r = 0; r < 8; ++r) {
      s1[r] += val[t][r];
      s2[r] += val[t][r] * val[t][r];
    }
#pragma unroll
  for (int mask = 1; mask < 16; mask <<= 1) {
#pragma unroll
    for (int r = 0; r < 8; ++r) {
      s1[r] += __shfl_xor(s1[r], mask, 32);
      s2[r] += __shfl_xor(s2[r], mask, 32);
    }
  }
  float mu[8], rstd[8];
#pragma unroll
  for (int r = 0; r < 8; ++r) {
    mu[r] = s1[r] * (1.f / HDIM);
    const float var = s2[r] * (1.f / HDIM) - mu[r] * mu[r];
    rstd[r] = rsqrtf(var + LN_EPS);
  }
#pragma unroll
  for (int t = 0; t < 8; ++t) {
    const int n = 16 * t + lo;
    const float g = lns[n];
    const float be = lnb[n];
#pragma unroll
    for (int r = 0; r < 8; ++r) {
      const int m = r + 8 * hi;
      const int node = base + m;
      if (node < nNodes)
        ffout[(size_t)node * HDIM + n] = (val[t][r] - mu[r]) * rstd[r] * g + be;
    }
  }
}

// ---- GCN layer: out = LN( relu((agg*invdeg)@W^T + b) + ff ) ----------------
__global__ __launch_bounds__(WAVES_PER_BLOCK * 32)
void gcn_layer_kernel(const float* __restrict__ agg,
                      const float* __restrict__ invdeg,
                      const float* __restrict__ W, const float* __restrict__ b,
                      const float* __restrict__ lns, const float* __restrict__ lnb,
                      const float* __restrict__ ff, float* __restrict__ out,
                      int nNodes) {
  __shared__ float wl[HDIM * LDSW];
  __shared__ float stage[WAVES_PER_BLOCK * ROWS_PER_WAVE * LDSW];

  const int tid = threadIdx.x;
  const int lane = tid & 31;
  const int wv = tid >> 5;
  const int lo = lane & 15;
  const int hi = lane >> 4;
  float* ast = stage + wv * ROWS_PER_WAVE * LDSW;
  const int base = blockIdx.x * ROWS_PER_BLOCK + wv * ROWS_PER_WAVE;

  // stage agg rows scaled by inv-degree (mean aggregation)
  for (int idx = lane; idx < ROWS_PER_WAVE * (HDIM / 4); idx += 32) {
    const int r = idx >> 5;
    const int c4 = idx & 31;
    const int node = base + r;
    float4 v = make_float4(0.f, 0.f, 0.f, 0.f);
    if (node < nNodes) {
      v = reinterpret_cast<const float4*>(agg + (size_t)node * HDIM)[c4];
      const float sc = invdeg[node];
      v.x *= sc; v.y *= sc; v.z *= sc; v.w *= sc;
    }
    float* p = ast + r * LDSW + c4 * 4;
    p[0] = v.x; p[1] = v.y; p[2] = v.z; p[3] = v.w;
  }
  load_weight_lds(wl, W, tid, blockDim.x);
  __syncthreads();

  v8f acc[8];
  gemm16x128(ast, wl, lane, acc);          // agg_mean @ W^T

  float val[8][8];
#pragma unroll
  for (int t = 0; t < 8; ++t) {
    const int n = 16 * t + lo;
    const float bb = b[n];
#pragma unroll
    for (int r = 0; r < 8; ++r) {
      const int m = r + 8 * hi;
      const int node = base + m;
      const float att = fmaxf(acc[t][r] + bb, 0.f);
      const float fv = (node < nNodes) ? ff[(size_t)node * HDIM + n] : 0.f;
      val[t][r] = att + fv;
    }
  }
  float s1[8], s2[8];
#pragma unroll
  for (int r = 0; r < 8; ++r) { s1[r] = 0.f; s2[r] = 0.f; }
#pragma unroll
  for (int t = 0; t < 8; ++t)
#pragma unroll
    for (int r = 0; r < 8; ++r) {
      s1[r] += val[t][r];
      s2[r] += val[t][r] * val[t][r];
    }
#pragma unroll
  for (int mask = 1; mask < 16; mask <<= 1) {
#pragma unroll
    for (int r = 0; r < 8; ++r) {
      s1[r] += __shfl_xor(s1[r], mask, 32);
      s2[r] += __shfl_xor(s2[r], mask, 32);
    }
  }
  float mu[8], rstd[8];
#pragma unroll
  for (int r = 0; r < 8; ++r) {
    mu[r] = s1[r] * (1.f / HDIM);
    const float var = s2[r] * (1.f / HDIM) - mu[r] * mu[r];
    rstd[r] = rsqrtf(var + LN_EPS);
  }
#pragma unroll
  for (int t = 0; t < 8; ++t) {
    const int n = 16 * t + lo;
    const float g = lns[n];
    const float be = lnb[n];
#pragma unroll
    for (int r = 0; r < 8; ++r) {
      const int m = r + 8 * hi;
      const int node = base + m;
      if (node < nNodes)
        out[(size_t)node * HDIM + n] = (val[t][r] - mu[r]) * rstd[r] * g + be;
    }
  }
}

// ---- scatter: agg[dst] += ff[src], one wave per edge, float4 per lane ------
__global__ void scatter_kernel(const float* __restrict__ ff,
                               const int* __restrict__ src,
                               const int* __restrict__ dst,
                               float* __restrict__ agg, int nEdges) {
  const int lane = threadIdx.x & 31;
  const int e = blockIdx.x * (blockDim.x >> 5) + (threadIdx.x >> 5);
  if (e >= nEdges) return;
  const int s = src[e];
  const int d = dst[e];
  const float4 v = reinterpret_cast<const float4*>(ff + (size_t)s * HDIM)[lane];
  float* p = agg + (size_t)d * HDIM + lane * 4;
  atomicAdd(p + 0, v.x);
  atomicAdd(p + 1, v.y);
  atomicAdd(p + 2, v.z);
  atomicAdd(p + 3, v.w);
}

// ---- small helpers ---------------------------------------------------------
__global__ void zero_i32(int* __restrict__ p, int n) {
  const int i = blockIdx.x * blockDim.x + threadIdx.x;
  if (i < n) p[i] = 0;
}
__global__ void zero_f32(float* __restrict__ p, long long n) {
  const long long i = (long long)blockIdx.x * blockDim.x + threadIdx.x;
  if (i < n) p[i] = 0.f;
}
__global__ void count_deg(const int* __restrict__ dst, int* __restrict__ deg,
                          int nEdges) {
  const int i = blockIdx.x * blockDim.x + threadIdx.x;
  if (i < nEdges) atomicAdd(&deg[dst[i]], 1);
}
__global__ void inv_deg_kernel(const int* __restrict__ deg,
                               float* __restrict__ inv, int n) {
  const int i = blockIdx.x * blockDim.x + threadIdx.x;
  if (i < n) {
    const int d = deg[i];
    inv[i] = (d > 0) ? (1.f / (float)d) : 0.f;
  }
}

// ---------------------------------------------------------------------------
extern "C" void kernel_launch(void* const* d_in, const int* in_sizes, int n_in,
                              void* d_out, int out_size, void* d_ws,
                              size_t ws_size, hipStream_t stream) {
  const float* features = (const float*)d_in[0];
  const float* ff_w1 = (const float*)d_in[1];
  const float* ff_b1 = (const float*)d_in[2];
  const float* ff_w2 = (const float*)d_in[3];
  const float* ff_b2 = (const float*)d_in[4];
  const float* ffln_s = (const float*)d_in[5];
  const float* ffln_b = (const float*)d_in[6];
  const float* gcn_w = (const float*)d_in[7];
  const float* gcn_b = (const float*)d_in[8];
  const float* ln_s = (const float*)d_in[9];
  const float* ln_b = (const float*)d_in[10];
  const int* src = (const int*)d_in[11];
  const int* dst = (const int*)d_in[12];
  (void)n_in; (void)out_size; (void)ws_size;

  const int nNodes = in_sizes[0] / HDIM;
  const int nEdges = in_sizes[11];

  // workspace carve-out (256B aligned): deg(int) | invdeg | ff | agg
  char* ws = (char*)d_ws;
  size_t off = 0;
  auto carve = [&](size_t bytes) {
    void* p = ws + off;
    off = (off + bytes + 255) & ~(size_t)255;
    return p;
  };
  int* degi = (int*)carve((size_t)nNodes * sizeof(int));
  float* invdeg = (float*)carve((size_t)nNodes * sizeof(float));
  float* ffbuf = (float*)carve((size_t)nNodes * HDIM * sizeof(float));
  float* aggbuf = (float*)carve((size_t)nNodes * HDIM * sizeof(float));

  // degree / inverse degree (shared by both layers)
  zero_i32<<<(nNodes + 255) / 256, 256, 0, stream>>>(degi, nNodes);
  count_deg<<<(nEdges + 255) / 256, 256, 0, stream>>>(dst, degi, nEdges);
  inv_deg_kernel<<<(nNodes + 255) / 256, 256, 0, stream>>>(degi, invdeg, nNodes);

  const int gemmBlocks = (nNodes + ROWS_PER_BLOCK - 1) / ROWS_PER_BLOCK;
  const long long featElems = (long long)nNodes * HDIM;
  float* layerOut = (float*)d_out;
  const float* X = features;
  for (int i = 0; i < 2; ++i) {
    const size_t wo = (size_t)i * HDIM * HDIM;
    const size_t bo = (size_t)i * HDIM;
    ff_layer_kernel<<<gemmBlocks, WAVES_PER_BLOCK * 32, 0, stream>>>(
        X, ff_w1 + wo, ff_b1 + bo, ff_w2 + wo, ff_b2 + bo, ffln_s + bo,
        ffln_b + bo, ffbuf, nNodes);
    zero_f32<<<(int)((featElems + 255) / 256), 256, 0, stream>>>(aggbuf,
                                                                 featElems);
    scatter_kernel<<<(nEdges + 7) / 8, 256, 0, stream>>>(ffbuf, src, dst,
                                                         aggbuf, nEdges);
    gcn_layer_kernel<<<gemmBlocks, WAVES_PER_BLOCK * 32, 0, stream>>>(
        aggbuf, invdeg, gcn_w + wo, gcn_b + bo, ln_s + bo, ln_b + bo, ffbuf,
        layerOut, nNodes);
    X = layerOut;
  }
}